// VolumeRotationSE3_85856396247752
// MI455X (gfx1250) — compile-verified
//
#include <hip/hip_runtime.h>

typedef float v2f __attribute__((ext_vector_type(2)));
typedef float v8f __attribute__((ext_vector_type(8)));

__device__ __forceinline__ float lane_gather(float v, int srcLane) {
    // ds_bpermute_b32: every lane reads `v` from lane `srcLane`.
    // Must be called with full EXEC (no divergence) so source lanes are active.
    return __int_as_float(__builtin_amdgcn_ds_bpermute(srcLane << 2, __float_as_int(v)));
}

__global__ __launch_bounds__(256)
void se3_rotate_kernel(const float* __restrict__ vol,
                       const float* __restrict__ R,
                       float* __restrict__ out) {
    constexpr int C   = 14;
    constexpr int DHW = 64 * 64 * 64;   // 262144

    // 1024 blocks per batch, 256 threads each -> one thread per voxel
    const int b  = blockIdx.x >> 10;
    const int sp = ((blockIdx.x & 1023) << 8) + threadIdx.x;
    const int w  = sp & 63;
    const int h  = (sp >> 6) & 63;
    const int d  = sp >> 12;

    // R[b] (b is uniform per block -> scalar loads)
    const float* Rb = R + b * 9;
    const float r00 = Rb[0], r01 = Rb[1], r02 = Rb[2];
    const float r10 = Rb[3], r11 = Rb[4], r12 = Rb[5];
    const float r20 = Rb[6], r21 = Rb[7], r22 = Rb[8];

    const float x = fmaf((float)w, 2.0f / 63.0f, -1.0f);
    const float y = fmaf((float)h, 2.0f / 63.0f, -1.0f);
    const float z = fmaf((float)d, 2.0f / 63.0f, -1.0f);

    // grid = A * [x,y,z,1], A[i][j] = R[2-j][2-i]
    const float gx = fmaf(r22, x, fmaf(r12, y, r02 * z));
    const float gy = fmaf(r21, x, fmaf(r11, y, r01 * z));
    const float gz = fmaf(r20, x, fmaf(r10, y, r00 * z));

    const float ix = (gx + 1.0f) * 31.5f;
    const float iy = (gy + 1.0f) * 31.5f;
    const float iz = (gz + 1.0f) * 31.5f;

    const float x0f = floorf(ix), y0f = floorf(iy), z0f = floorf(iz);
    const float tx = ix - x0f, ty = iy - y0f, tz = iz - z0f;
    const int x0 = (int)x0f, y0 = (int)y0f, z0 = (int)z0f;

    float s[C];
    #pragma unroll
    for (int c = 0; c < C; ++c) s[c] = 0.0f;

    // sample box intersects the volume at all?
    const bool inter = (x0 >= -1) & (x0 <= 63) & (y0 >= -1) & (y0 <= 63) &
                       (z0 >= -1) & (z0 <= 63);
    if (inter) {
        // given inter: x0 in [-1,63] so x1=x0+1 in [0,64]
        const float wx0 = (x0 >= 0)  ? (1.0f - tx) : 0.0f;
        const float wx1 = (x0 <= 62) ? tx          : 0.0f;
        const float wy0 = (y0 >= 0)  ? (1.0f - ty) : 0.0f;
        const float wy1 = (y0 <= 62) ? ty          : 0.0f;
        const float wz0 = (z0 >= 0)  ? (1.0f - tz) : 0.0f;
        const float wz1 = (z0 <= 62) ? tz          : 0.0f;

        const int xc0 = (x0 < 0) ? 0 : x0;
        const int xc1 = (x0 >= 63) ? 63 : x0 + 1;
        const int yc0 = (y0 < 0) ? 0 : y0;
        const int yc1 = (y0 >= 63) ? 63 : y0 + 1;
        const int zc0 = (z0 < 0) ? 0 : z0;
        const int zc1 = (z0 >= 63) ? 63 : z0 + 1;

        const float wz0y0 = wz0 * wy0, wz0y1 = wz0 * wy1;
        const float wz1y0 = wz1 * wy0, wz1y1 = wz1 * wy1;
        const float w000 = wz0y0 * wx0, w001 = wz0y0 * wx1;
        const float w010 = wz0y1 * wx0, w011 = wz0y1 * wx1;
        const float w100 = wz1y0 * wx0, w101 = wz1y0 * wx1;
        const float w110 = wz1y1 * wx0, w111 = wz1y1 * wx1;

        const int zb0 = zc0 << 12, zb1 = zc1 << 12;
        const int yb0 = yc0 << 6,  yb1 = yc1 << 6;
        const int o000 = zb0 + yb0 + xc0, o001 = zb0 + yb0 + xc1;
        const int o010 = zb0 + yb1 + xc0, o011 = zb0 + yb1 + xc1;
        const int o100 = zb1 + yb0 + xc0, o101 = zb1 + yb0 + xc1;
        const int o110 = zb1 + yb1 + xc0, o111 = zb1 + yb1 + xc1;

        const float* vb = vol + (size_t)b * (C * DHW);
        #pragma unroll
        for (int c = 0; c < C; ++c) {
            const float* vc = vb + (size_t)c * DHW;
            float acc;
            acc = w000 * vc[o000];
            acc = fmaf(w001, vc[o001], acc);
            acc = fmaf(w010, vc[o010], acc);
            acc = fmaf(w011, vc[o011], acc);
            acc = fmaf(w100, vc[o100], acc);
            acc = fmaf(w101, vc[o101], acc);
            acc = fmaf(w110, vc[o110], acc);
            acc = fmaf(w111, vc[o111], acc);
            s[c] = acc;
        }
    }
    // control flow reconverges here: EXEC is all-ones again for WMMA/bpermute

    float* ob = out + (size_t)b * (C * DHW);

    // scalar (l=0) channels pass through
    ob[0 * DHW + sp] = s[0];
    ob[1 * DHW + sp] = s[1];

    // ----- l=1 channel mixing via V_WMMA_F32_16X16X4_F32 -----
    // D(16x16) = A(16x4) x B(4x16); A top-left 3x3 = R[b], rest zero
    // => D rows 3..15 are zero. Columns of B = 16 consecutive voxels of this
    // wave; two WMMAs per group cover the wave's 32 voxels.
    const int  lane = threadIdx.x & 31;
    const int  m    = lane & 15;
    const bool hi   = lane >= 16;

    // A layout (32-bit 16x4): VGPR0 = K0 (lanes 0-15, M=lane) / K2 (lanes 16-31)
    //                         VGPR1 = K1 (lanes 0-15)         / K3 (zero)
    v2f Aop;
    Aop.x = (m == 0) ? (hi ? r02 : r00)
          : (m == 1) ? (hi ? r12 : r10)
          : (m == 2) ? (hi ? r22 : r20) : 0.0f;
    Aop.y = (!hi) ? ((m == 0) ? r01 : (m == 1) ? r11 : (m == 2) ? r21 : 0.0f)
                  : 0.0f;

    #pragma unroll
    for (int g = 0; g < 4; ++g) {
        const float s0 = s[2 + 3 * g];
        const float s1 = s[3 + 3 * g];
        const float s2 = s[4 + 3 * g];

        // B layout (32-bit 4x16): VGPR0 = row K0 (lanes 0-15) / K2 (lanes 16-31)
        //                         VGPR1 = row K1 (lanes 0-15) / K3 (zero)
        // All lane_gathers run under full EXEC, then blend with cndmask.
        const float t2 = lane_gather(s2, m);        // hi lanes read lane-16
        const float t0 = lane_gather(s0, m | 16);   // lo lanes read lane+16
        const float t1 = lane_gather(s1, m | 16);

        v2f b0, b1;
        b0.x = hi ? t2 : s0;        // half 0: voxels waveBase+0..15
        b0.y = hi ? 0.0f : s1;
        b1.x = hi ? s2 : t0;        // half 1: voxels waveBase+16..31
        b1.y = hi ? 0.0f : t1;

        v8f cz = {};
        v8f d0 = __builtin_amdgcn_wmma_f32_16x16x4_f32(
            false, Aop, false, b0, (short)0, cz, false, false);
        v8f d1 = __builtin_amdgcn_wmma_f32_16x16x4_f32(
            false, Aop, false, b1, (short)0, cz, false, false);

        // D rows 0..2 (lanes 0-15 of VGPR r) hold mixed channel (2+3g+r).
        // A hi lane's value sits in lane (lane-16) of d1[r]; gather it with
        // full EXEC, then select.
        #pragma unroll
        for (int r = 0; r < 3; ++r) {
            const float v0 = d0[r];
            const float tg = lane_gather(d1[r], m);   // unconditional: full EXEC
            const float mine = hi ? tg : v0;
            ob[(size_t)(2 + 3 * g + r) * DHW + sp] = mine;
        }
    }
}

extern "C" void kernel_launch(void* const* d_in, const int* in_sizes, int n_in,
                              void* d_out, int out_size, void* d_ws, size_t ws_size,
                              hipStream_t stream) {
    (void)in_sizes; (void)n_in; (void)out_size; (void)d_ws; (void)ws_size;
    const float* vol = (const float*)d_in[0];   // (8,14,64,64,64) f32
    const float* R   = (const float*)d_in[1];   // (8,3,3) f32
    float* out = (float*)d_out;                 // (8,14,64,64,64) f32

    dim3 block(256);
    dim3 grid(8 * 1024);                        // 1024 blocks per batch
    hipLaunchKernelGGL(se3_rotate_kernel, grid, block, 0, stream, vol, R, out);
}